// QMixer_3040836845938
// MI455X (gfx1250) — compile-verified
//
#include <hip/hip_runtime.h>

// ---------------------------------------------------------------------------
// QMixer for MI455X (gfx1250): one workgroup per graph component.
//  - fp32 WMMA (v_wmma_f32_16x16x4_f32) for all GEMMs (exact-precision path)
//  - 64x128 feature tiles resident in LDS; relation aggregation via ds_add_f32
//  - wave32 shuffle reductions for the final [128]->1 projections
// ---------------------------------------------------------------------------

typedef __attribute__((ext_vector_type(2))) float v2f;
typedef __attribute__((ext_vector_type(8))) float v8f;

#define N_PER 64          // nodes per graph
#define A_PER 16          // ally nodes per graph (first 16)
#define D     128         // feature dim (D_IN == D_EMB)
#define DEG   16          // edges per node
#define EPG   (N_PER*DEG) // 1024 edges per graph
#define NREL  3

// A fragment: 16x4 fp32 tile of row-major S (stride D) starting at (m0, k0).
// lanes 0-15 -> rows m0..m0+15, K = k0,k0+1 ; lanes 16-31 -> same rows, K = k0+2,k0+3
__device__ __forceinline__ v2f lda(const float* S, int m0, int k0, int lo, int hi) {
  const float* p = S + (m0 + lo) * D + k0 + 2 * hi;
  v2f a; a.x = p[0]; a.y = p[1];   // 8B-aligned pair -> ds_load_b64
  return a;
}

// B fragment: 4x16 fp32 tile of row-major W [D x D] at (k0, column 'col').
// lanes 0-15: K = k0,k0+1 ; lanes 16-31: K = k0+2,k0+3 (column fixed per lane)
__device__ __forceinline__ v2f ldb(const float* __restrict__ W, int k0, int col, int hi) {
  const int kk = k0 + 2 * hi;
  v2f b; b.x = W[kk * D + col]; b.y = W[(kk + 1) * D + col];
  return b;
}

// Dual-branch GEMM: accw += Sw @ Ww , accv += Sv @ Wv   (A from LDS, B from L2)
// Each wave computes a 64x16 output slab (4 M-tiles) per branch.
__device__ __forceinline__ void gemm_two(const float* Sw, const float* Sv,
                                         const float* __restrict__ Ww,
                                         const float* __restrict__ Wv,
                                         v8f* accw, v8f* accv,
                                         int lo, int hi, int col) {
#pragma unroll 4
  for (int k0 = 0; k0 < D; k0 += 4) {
    v2f bw = ldb(Ww, k0, col, hi);
    v2f bv = ldb(Wv, k0, col, hi);
#pragma unroll
    for (int m = 0; m < 4; ++m) {
      v2f aw = lda(Sw, m * 16, k0, lo, hi);
      v2f av = lda(Sv, m * 16, k0, lo, hi);
      accw[m] = __builtin_amdgcn_wmma_f32_16x16x4_f32(
          false, aw, false, bw, (short)0, accw[m], false, false);
      accv[m] = __builtin_amdgcn_wmma_f32_16x16x4_f32(
          false, av, false, bv, (short)0, accv[m], false, false);
    }
  }
}

__global__ __launch_bounds__(256)
void qmix_fused_kernel(const float* __restrict__ x,       // [N, D]
                       const float* __restrict__ qs,      // [NA, 1]
                       const float* __restrict__ Wself_w, // [D, D]
                       const float* __restrict__ Wrel_w,  // [R, D, D]
                       const float* __restrict__ bw,      // [D]
                       const float* __restrict__ W1w,     // [D, D]
                       const float* __restrict__ b1w,     // [D]
                       const float* __restrict__ W2w,     // [D, 1]
                       const float* __restrict__ b2w,     // [1]
                       const float* __restrict__ Wself_v,
                       const float* __restrict__ Wrel_v,
                       const float* __restrict__ bv,
                       const float* __restrict__ W1v,
                       const float* __restrict__ b1v,
                       const float* __restrict__ W2v,
                       const float* __restrict__ b2v,
                       const int* __restrict__ src,       // [E]
                       const int* __restrict__ dst,       // [E]
                       const int* __restrict__ etype,     // [E]
                       float* __restrict__ out)           // [B]
{
  __shared__ float xs[N_PER * D];   // 32 KB: x tile, later h_w, later reduce scratch
  __shared__ float buf[N_PER * D];  // 32 KB: relation aggregate, later h_v

  const int g    = blockIdx.x;
  const int tid  = threadIdx.x;
  const int lane = tid & 31;
  const int wave = tid >> 5;        // 8 waves
  const int lo   = lane & 15;
  const int hi   = lane >> 4;
  const int col  = wave * 16 + lo;  // this lane's output column (0..127)

  // ---- Phase A: stage x[g] into LDS (coalesced float4) -------------------
  {
    const float4* xg4 = (const float4*)(x + (size_t)g * N_PER * D);
    float4* xs4 = (float4*)xs;
    for (int i = tid; i < N_PER * D / 4; i += 256) xs4[i] = xg4[i];
  }
  __syncthreads();

  // ---- Phase B: GNN layer, both branches ---------------------------------
  v8f hw[4], hv[4];
  {
    const float bwv = bw[col];
    const float bvv = bv[col];
#pragma unroll
    for (int m = 0; m < 4; ++m)
#pragma unroll
      for (int j = 0; j < 8; ++j) { hw[m][j] = bwv; hv[m][j] = bvv; }
  }

  // self term: x @ Wself_{w,v}
  gemm_two(xs, xs, Wself_w, Wself_v, hw, hv, lo, hi, col);

  // relation terms: build agg_r in LDS, then agg_r @ Wrel_{w,v}[r]
  const int eb = g * EPG;
  const int f  = tid & 127;          // feature lane for edge scatter
  for (int r = 0; r < NREL; ++r) {
    __syncthreads();                 // prior GEMM done reading buf
    {
      float4 z = make_float4(0.f, 0.f, 0.f, 0.f);
      float4* b4 = (float4*)buf;
      for (int i = tid; i < N_PER * D / 4; i += 256) b4[i] = z;
    }
    __syncthreads();
    // scatter: 2 edges per iteration, 128 feature lanes each; LDS f32 atomics
    for (int base = tid >> 7; base < EPG; base += 2) {
      const int e = eb + base;
      if (etype[e] == r) {
        const int sl = src[e] - g * N_PER;
        const int dl = dst[e] - g * N_PER;
        atomicAdd(&buf[dl * D + f], xs[sl * D + f]);
      }
    }
    __syncthreads();
    gemm_two(buf, buf, Wrel_w + r * D * D, Wrel_v + r * D * D,
             hw, hv, lo, hi, col);
  }
  __syncthreads();                   // all waves done with xs/buf contents

  // ---- Phase C: ReLU, park h_w in xs and h_v in buf ----------------------
#pragma unroll
  for (int m = 0; m < 4; ++m)
#pragma unroll
    for (int j = 0; j < 8; ++j) {
      const int row = m * 16 + hi * 8 + j;     // C-layout row for this element
      xs[row * D + col]  = fmaxf(hw[m][j], 0.f);
      buf[row * D + col] = fmaxf(hv[m][j], 0.f);
    }
  __syncthreads();

  // ---- Phase D: FF1  t = h @ W1 + b1  (both branches) --------------------
  v8f tw[4], tv[4];
  {
    const float b1wv = b1w[col];
    const float b1vv = b1v[col];
#pragma unroll
    for (int m = 0; m < 4; ++m)
#pragma unroll
      for (int j = 0; j < 8; ++j) { tw[m][j] = b1wv; tv[m][j] = b1vv; }
  }
  gemm_two(xs, buf, W1w, W1v, tw, tv, lo, hi, col);
  __syncthreads();                   // h tiles no longer needed

  // ---- Phase E: FF2 (relu(t) @ W2) + per-graph reductions ----------------
  // reuse freed xs[] head as scratch: wsum[64], vsum[64]
  float* wsum = xs;
  float* vsum = xs + 64;
  if (tid < 128) xs[tid] = 0.f;
  __syncthreads();

  const float w2wv = W2w[col];
  const float w2vv = W2v[col];
#pragma unroll
  for (int m = 0; m < 4; ++m) {
#pragma unroll
    for (int j = 0; j < 8; ++j) {
      float pw = fmaxf(tw[m][j], 0.f) * w2wv;
      float pv = fmaxf(tv[m][j], 0.f) * w2vv;
#pragma unroll
      for (int off = 8; off; off >>= 1) {      // reduce 16 columns per half-wave
        pw += __shfl_xor(pw, off, 16);
        pv += __shfl_xor(pv, off, 16);
      }
      if (lo == 0) {                            // lanes 0 and 16 hold row sums
        const int row = m * 16 + hi * 8 + j;
        atomicAdd(&wsum[row], pw);
        atomicAdd(&vsum[row], pv);
      }
    }
  }
  __syncthreads();

  if (wave == 0) {
    // v_sum = sum_m (v[m] + b2v); q_tot = sum_{m<16} (w[m] + b2w) * qs[g,m]
    float t = vsum[lane] + vsum[lane + 32] + 2.0f * b2v[0];
    if (lane < A_PER)
      t += (wsum[lane] + b2w[0]) * qs[g * A_PER + lane];
#pragma unroll
    for (int off = 16; off; off >>= 1) t += __shfl_xor(t, off, 32);
    if (lane == 0) out[g] = t;
  }
}

extern "C" void kernel_launch(void* const* d_in, const int* in_sizes, int n_in,
                              void* d_out, int out_size, void* d_ws, size_t ws_size,
                              hipStream_t stream) {
  (void)in_sizes; (void)n_in; (void)d_ws; (void)ws_size;
  const int B = out_size;  // one graph -> one output scalar
  qmix_fused_kernel<<<B, 256, 0, stream>>>(
      (const float*)d_in[0],  (const float*)d_in[1],
      (const float*)d_in[2],  (const float*)d_in[3],  (const float*)d_in[4],
      (const float*)d_in[5],  (const float*)d_in[6],  (const float*)d_in[7],
      (const float*)d_in[8],
      (const float*)d_in[9],  (const float*)d_in[10], (const float*)d_in[11],
      (const float*)d_in[12], (const float*)d_in[13], (const float*)d_in[14],
      (const float*)d_in[15],
      (const int*)d_in[16],   (const int*)d_in[17],   (const int*)d_in[18],
      (float*)d_out);
}